// DVSA_44332652430063
// MI455X (gfx1250) — compile-verified
//
#include <hip/hip_runtime.h>

typedef __attribute__((ext_vector_type(2))) float v2f;
typedef __attribute__((ext_vector_type(8))) float v8f;

#define NA 16
#define NS 64
#define NB 100
#define NE 12
#define MAXLEN 20
#define DD 512
#define NCOL (NA * NE)   /* 192 */
#define NROW (NA * NS)   /* 1024 */
#define MROW (NA * NS * NB) /* 102400 */
#define EPSF 1e-5f
#define DELTAF 0.2f

__device__ __forceinline__ float relu_f(float x) { return x > 0.f ? x : 0.f; }

// ---------------- kernel 0: zero global accumulators ----------------
__global__ void k_init(float* gsum, unsigned int* gcnt) {
  if (threadIdx.x == 0) { *gsum = 0.f; *gcnt = 0u; }
}

// ---------------- kernel 1: IoU d_ti_n (Na,Ns,Nb,MAXLEN) ----------------
__global__ void k_iou(const float* __restrict__ boxes,
                      const float* __restrict__ dboxes,
                      float* __restrict__ out) {
  int idx = blockIdx.x * blockDim.x + threadIdx.x;
  const int total = NA * NS * NB * MAXLEN;
  if (idx >= total) return;
  int n = idx % MAXLEN;
  int bi = idx / MAXLEN;                 // (a*NS+s)*NB + b
  int a = bi / (NS * NB);
  const float* A = boxes + (size_t)bi * 4;
  const float* B = dboxes + ((size_t)a * MAXLEN + n) * 4;
  float iw = fminf(A[2], B[2]) - fmaxf(A[0], B[0]);
  float ih = fminf(A[3], B[3]) - fmaxf(A[1], B[1]);
  bool pos = (iw > 0.f) && (ih > 0.f);
  float inter = pos ? iw * ih : 0.f;
  float a1 = (A[2] - A[0]) * (A[3] - A[1]);
  float a2 = (B[2] - B[0]) * (B[3] - B[1]);
  float denom = a1 + a2 - inter;
  out[idx] = pos ? inter / (denom > 0.f ? denom : 1.0f) : 0.f;
}

// ---------------- kernel 2: S_ = vis @ word^T, masked (WMMA f32 16x16x4) ----
// 1 wave = one 16x16 tile of C. Grid: 6400*12 tiles / 8 waves per block.
__global__ void k_gemm(const float* __restrict__ vis,
                       const float* __restrict__ word,
                       const int* __restrict__ entlen,
                       float* __restrict__ S) {
  int wave = threadIdx.x >> 5;
  int lane = threadIdx.x & 31;
  int tile = blockIdx.x * 8 + wave;            // 0 .. 76799
  const int tilesN = NCOL / 16;                // 12
  int tm = tile / tilesN;                      // 0 .. 6399
  int tn = tile % tilesN;
  int l15 = lane & 15;
  int kh  = (lane >> 4) * 2;                   // 0 or 2
  const float* arow = vis  + (size_t)(tm * 16 + l15) * DD + kh;
  const float* brow = word + (size_t)(tn * 16 + l15) * DD + kh;
  v8f c = {};
  for (int k = 0; k < DD; k += 4) {
    v2f av, bv;
    av[0] = arow[k]; av[1] = arow[k + 1];
    bv[0] = brow[k]; bv[1] = brow[k + 1];
    c = __builtin_amdgcn_wmma_f32_16x16x4_f32(false, av, false, bv,
                                              (short)0, c, false, false);
  }
  int n = tn * 16 + l15;                       // global column
  int aw = n / NE, e = n % NE;
  float keep = (e < entlen[aw]) ? 1.f : 0.f;   // word mask -> zero column
  int mbase = tm * 16 + 8 * (lane >> 4);       // C layout: M = r + 8*(lane>=16)
  #pragma unroll
  for (int r = 0; r < 8; ++r)
    S[(size_t)(mbase + r) * NCOL + n] = c[r] * keep;
}

// ---------------- kernel 3: max/argmax over Nb -> D_sim, D_ind -------------
__global__ void k_reduce(const float* __restrict__ S,
                         float* __restrict__ dind,
                         float* __restrict__ dsim) {
  int idx = blockIdx.x * blockDim.x + threadIdx.x;
  if (idx >= NROW * NCOL) return;
  int r = idx / NCOL, c = idx % NCOL;
  const float* p = S + (size_t)r * NB * NCOL + c;
  float best = p[0];
  int bi = 0;
  for (int b = 1; b < NB; ++b) {
    float v = p[(size_t)b * NCOL];
    if (v > best) { best = v; bi = b; }        // strict > : first-max like argmax
  }
  dind[idx] = (float)bi;
  dsim[idx] = best;
}

// ---------------- kernel 4: Sf[a][s][a2] from normalized S -----------------
__global__ void k_sf(const float* __restrict__ dsim,
                     const int* __restrict__ entlen,
                     float* __restrict__ Sf) {
  __shared__ float sf[NS * NA];
  int tid = threadIdx.x;                       // 0..191
  int a = blockIdx.x;
  for (int i = tid; i < NS * NA; i += NCOL) sf[i] = 0.f;
  __syncthreads();
  const float* base = dsim + (size_t)a * NS * NCOL + tid;
  float mn = 1e30f, mx = -1e30f;
  for (int s = 0; s < NS; ++s) {
    float v = base[s * NCOL];
    mn = fminf(mn, v); mx = fmaxf(mx, v);
  }
  int a2 = tid / NE;
  int len = entlen[a2];
  float div = (float)(len > 1 ? len : 1);
  float inv = 1.0f / (mx - mn + EPSF);
  for (int s = 0; s < NS; ++s) {
    float v = base[s * NCOL];
    float sn = v * (v - mn) * inv;             // S * (S - mn)/(mx - mn + EPS)
    atomicAdd(&sf[s * NA + a2], sn / div);
  }
  __syncthreads();
  for (int i = tid; i < NS * NA; i += NCOL)
    Sf[(size_t)a * NS * NA + i] = sf[i];
}

// ---------------- kernel 5: vis_loss gram per (a,e) (WMMA f32 16x16x4) -----
__global__ void k_vis(const float* __restrict__ vis,
                      const float* __restrict__ dsim,
                      const float* __restrict__ dind,
                      const int* __restrict__ entlen,
                      float* __restrict__ gsum,
                      unsigned int* __restrict__ gcnt) {
  const int KS = 128;                          // K slice per LDS pass
  __shared__ float Vs[NS][KS + 4];             // pad->132: bank-conflict free
  __shared__ float sims[NS];
  __shared__ int   rows[NS];
  __shared__ float red[NS][4];
  __shared__ float mnv, mxv, bsum;
  __shared__ unsigned int bcnt;

  int a = blockIdx.x / NE;
  int e = blockIdx.x % NE;
  if (e >= entlen[a]) return;                  // masked entity: block-uniform

  int tid = threadIdx.x;                       // 0..255
  if (tid == 0) { bsum = 0.f; bcnt = 0u; }
  if (tid < NS) {
    int r = a * NS + tid;
    int col = a * NE + e;
    sims[tid] = dsim[(size_t)r * NCOL + col];
    rows[tid] = r * NB + (int)dind[(size_t)r * NCOL + col];
  }
  __syncthreads();
  if (tid == 0) {
    float mn = 1e30f, mx = -1e30f;
    for (int s = 0; s < NS; ++s) { mn = fminf(mn, sims[s]); mx = fmaxf(mx, sims[s]); }
    mnv = mn; mxv = mx;
  }
  __syncthreads();
  {                                            // partial sum of squares
    int s = tid >> 2, q = tid & 3;
    const float* p = vis + (size_t)rows[s] * DD + q * 128;
    float acc = 0.f;
    for (int k = 0; k < 128; ++k) { float x = p[k]; acc += x * x; }
    red[s][q] = acc;
  }
  __syncthreads();
  if (tid < NS) {
    float nrm = sqrtf(red[tid][0] + red[tid][1] + red[tid][2] + red[tid][3]) + EPSF;
    float simn = (sims[tid] - mnv) / (mxv - mnv + EPSF);
    sims[tid] = simn / nrm;                    // combined row scale
  }
  __syncthreads();

  int wave = tid >> 5, lane = tid & 31, l15 = lane & 15, kh = (lane >> 4) * 2;
  int t0 = wave * 2, t1 = wave * 2 + 1;        // two 16x16 tiles per wave
  int m0 = (t0 >> 2) * 16 + l15, n0 = (t0 & 3) * 16 + l15;
  int m1 = (t1 >> 2) * 16 + l15, n1 = (t1 & 3) * 16 + l15;
  v8f acc0 = {}, acc1 = {};

  for (int p = 0; p < DD / KS; ++p) {
    for (int i = 0; i < (NS * KS) / 256; ++i) {        // stage V slice
      int l = tid + i * 256;
      int s = l >> 7, k = l & (KS - 1);
      Vs[s][k] = vis[(size_t)rows[s] * DD + p * KS + k] * sims[s];
    }
    __syncthreads();
    for (int k = 0; k < KS; k += 4) {
      v2f av, bv;
      av[0] = Vs[m0][k + kh]; av[1] = Vs[m0][k + kh + 1];
      bv[0] = Vs[n0][k + kh]; bv[1] = Vs[n0][k + kh + 1];
      acc0 = __builtin_amdgcn_wmma_f32_16x16x4_f32(false, av, false, bv,
                                                   (short)0, acc0, false, false);
      av[0] = Vs[m1][k + kh]; av[1] = Vs[m1][k + kh + 1];
      bv[0] = Vs[n1][k + kh]; bv[1] = Vs[n1][k + kh + 1];
      acc1 = __builtin_amdgcn_wmma_f32_16x16x4_f32(false, av, false, bv,
                                                   (short)0, acc1, false, false);
    }
    __syncthreads();
  }

  float lsum = 0.f;
  unsigned int lcnt = 0u;
  int half = 8 * (lane >> 4);
  #pragma unroll
  for (int r = 0; r < 8; ++r) {
    int mg0 = (t0 >> 2) * 16 + r + half, ng0 = (t0 & 3) * 16 + l15;
    if (mg0 != ng0) { float g = 1.0f - acc0[r]; if (g != 0.0f) { lsum += g; lcnt++; } }
    int mg1 = (t1 >> 2) * 16 + r + half, ng1 = (t1 & 3) * 16 + l15;
    if (mg1 != ng1) { float g = 1.0f - acc1[r]; if (g != 0.0f) { lsum += g; lcnt++; } }
  }
  atomicAdd(&bsum, lsum);
  atomicAdd(&bcnt, lcnt);
  __syncthreads();
  if (tid == 0) { atomicAdd(gsum, bsum); atomicAdd(gcnt, bcnt); }
}

// ---------------- kernel 6: frame_score mean + margin_loss -----------------
__global__ void k_final(const float* __restrict__ Sf,
                        const float* __restrict__ gsum,
                        const unsigned int* __restrict__ gcnt,
                        float* __restrict__ out_loss) {
  __shared__ float red[1024];
  int tid = threadIdx.x;
  int x = tid >> 6, s = tid & 63;
  float sfd = Sf[(size_t)x * NS * NA + s * NA + x];
  float t = 0.f;
  for (int i = 0; i < NA; ++i) {
    t += relu_f(Sf[(size_t)i * NS * NA + s * NA + x] - sfd + DELTAF); // term1
    t += relu_f(Sf[(size_t)x * NS * NA + s * NA + i] - sfd + DELTAF); // term2
  }
  red[tid] = t * (1.0f / (float)NA);
  __syncthreads();
  for (int off = 512; off > 0; off >>= 1) {
    if (tid < off) red[tid] += red[tid + off];
    __syncthreads();
  }
  if (tid == 0) {
    float frame_mean = red[0] / (float)(NA * NS);
    unsigned int c = *gcnt;
    float vis_loss = *gsum / (float)(c > 0u ? c : 1u);
    out_loss[0] = (frame_mean + 1.0f * vis_loss) * 10.0f;
  }
}

extern "C" void kernel_launch(void* const* d_in, const int* in_sizes, int n_in,
                              void* d_out, int out_size, void* d_ws, size_t ws_size,
                              hipStream_t stream) {
  const float* vis    = (const float*)d_in[0];   // (102400, 512)
  const float* word   = (const float*)d_in[1];   // (192, 512)
  const float* boxes  = (const float*)d_in[2];   // (16,64,100,4)
  const float* dboxes = (const float*)d_in[3];   // (16,20,4)
  const int*   entlen = (const int*)d_in[4];     // (16,)

  float* out_ind  = (float*)d_out;               // 196608
  float* out_sim  = out_ind + NROW * NCOL;       // 196608
  float* out_loss = out_sim + NROW * NCOL;       // 1
  float* out_iou  = out_loss + 1;                // 2048000

  float* Sws  = (float*)d_ws;                    // 102400*192 floats (~75 MB)
  float* Sfws = Sws + (size_t)MROW * NCOL;       // 16*64*16 floats
  float* gsum = Sfws + NA * NS * NA;
  unsigned int* gcnt = (unsigned int*)(gsum + 1);

  k_init  <<<1, 32, 0, stream>>>(gsum, gcnt);
  k_iou   <<<(NA*NS*NB*MAXLEN + 255) / 256, 256, 0, stream>>>(boxes, dboxes, out_iou);
  k_gemm  <<<(MROW / 16) * (NCOL / 16) / 8, 256, 0, stream>>>(vis, word, entlen, Sws);
  k_reduce<<<(NROW * NCOL + 255) / 256, 256, 0, stream>>>(Sws, out_ind, out_sim);
  k_sf    <<<NA, NCOL, 0, stream>>>(out_sim, entlen, Sfws);
  k_vis   <<<NA * NE, 256, 0, stream>>>(vis, out_sim, out_ind, entlen, gsum, gcnt);
  k_final <<<1, 1024, 0, stream>>>(Sfws, gsum, gcnt, out_loss);
}